// RPN_29111288333008
// MI455X (gfx1250) — compile-verified
//
#include <hip/hip_runtime.h>
#include <stdint.h>

#define M_PROP   10000
#define TOPK_OUT 2000
#define NMS_THR  0.7f
#define NSORT    16384          // next pow2 >= M_PROP
#define NTHREADS 1024           // 32 wave32 waves
#define NW       313            // ceil(M_PROP/32) keep-mask words == #windows
#define TAILMASK 0xFFFFu        // valid bits in word 312 (10000 - 312*32 = 16)

// Workspace layout (d_ws):
//   [0,      160000) : float4 sorted boxes (x1,y1,x2,y2), score-descending order
//   [160000, 200000) : uint32 sorted original indices
#define WS_ORDER_OFF 160000

// LDS layout for the NMS kernel (dynamic shared):
#define L_SBOX   0                    // float4[10000]            160000 B
#define L_KEEP   160000               // uint[316]                  1264 B
#define L_WSUM   161264               // uint[316]                  1264 B
#define L_LBOX   162528               // float4[32]                  512 B (16-aligned)
#define L_LAREA  163040               // float[32]                   128 B
#define L_MISC   163168               // uint[4]                      16 B
#define L_TOTAL  163200

// LDS byte offset of a generic pointer known to be in LDS (addrspacecast -> AS3).
typedef const char __attribute__((address_space(3))) *lds_cptr_t;
__device__ __forceinline__ unsigned lds_off_u32(const void* p) {
    return (unsigned)(unsigned long long)(lds_cptr_t)p;
}

// ---------------------------------------------------------------------------
// Kernel A: stable descending sort of (score, index) via bitonic sort in LDS.
// Single workgroup; 16384 x u64 keys = 128 KB dynamic LDS.
// ---------------------------------------------------------------------------
__global__ void rpn_sort_kernel(const float* __restrict__ prop,
                                float4* __restrict__ sboxes,
                                unsigned* __restrict__ order) {
    extern __shared__ unsigned long long skey[];
    const int tid = threadIdx.x;

    for (int t = tid; t < NSORT; t += NTHREADS) {
        unsigned long long k;
        if (t < M_PROP) {
            unsigned b = __float_as_uint(prop[(size_t)t * 6 + 5]);
            unsigned u = (b & 0x80000000u) ? ~b : (b | 0x80000000u); // order-preserving
            k = ((unsigned long long)(~u) << 32) | (unsigned)t;      // asc key = desc score
        } else {
            k = (0xFFFFFFFFull << 32) | (unsigned)t;                 // pad sorts last
        }
        skey[t] = k;
    }
    __syncthreads();

    for (unsigned kk = 2; kk <= NSORT; kk <<= 1) {
        for (unsigned j = kk >> 1; j > 0; j >>= 1) {
            for (unsigned t = tid; t < NSORT; t += NTHREADS) {
                unsigned ixj = t ^ j;
                if (ixj > t) {
                    unsigned long long a = skey[t], b = skey[ixj];
                    bool asc = ((t & kk) == 0);
                    if ((a > b) == asc) { skey[t] = b; skey[ixj] = a; }
                }
            }
            __syncthreads();
        }
    }

    for (int r = tid; r < M_PROP; r += NTHREADS) {
        unsigned idx = (unsigned)(skey[r] & 0xFFFFFFFFull);
        order[r] = idx;
        const float* p = prop + (size_t)idx * 6;
        sboxes[r] = make_float4(p[1], p[2], p[3], p[4]);
    }
}

__device__ __forceinline__ float iou_f(float ax1, float ay1, float ax2, float ay2,
                                       float areaa, const float4 b) {
    float iw = fminf(ax2, b.z) - fmaxf(ax1, b.x);
    float ih = fminf(ay2, b.w) - fmaxf(ay1, b.y);
    iw = fmaxf(iw, 0.0f); ih = fmaxf(ih, 0.0f);
    float inter = iw * ih;
    float areab = (b.z - b.x) * (b.w - b.y);
    return inter / (areaa + areab - inter + 1e-8f);
}

// ---------------------------------------------------------------------------
// Kernel B: blocked greedy NMS, box table LDS-resident via CDNA5 async DMA.
//   Window w == keep word w (32 boxes). Per window:
//     (a) wave 0 resolves intra-window greedy serially (ballot/shfl, no barriers)
//         and compacts alive leaders into LDS.
//     (b) all 32 waves apply the leaders to every later word; one wave owns one
//         word, suppression combined via __ballot, lane 0 stores -- no atomics.
//   2 workgroup barriers per window (626 total vs 10000 in the naive loop).
// ---------------------------------------------------------------------------
__global__ void rpn_nms_kernel(const float* __restrict__ prop,
                               const float4* __restrict__ sboxes,
                               const unsigned* __restrict__ order,
                               float* __restrict__ out) {
    extern __shared__ char smem[];
    float4*   sbox  = (float4*)(smem + L_SBOX);
    unsigned* keepw = (unsigned*)(smem + L_KEEP);
    unsigned* wsum  = (unsigned*)(smem + L_WSUM);
    float4*   lbox  = (float4*)(smem + L_LBOX);
    float*    larea = (float*)(smem + L_LAREA);
    unsigned* misc  = (unsigned*)(smem + L_MISC);
    const int tid   = threadIdx.x;
    const int lane  = tid & 31;
    const int wave  = tid >> 5;

    // CDNA5 async global->LDS bulk fill (ASYNCcnt path).
    for (int r = tid; r < M_PROP; r += NTHREADS) {
        unsigned dst = lds_off_u32(&sbox[r]);
        unsigned long long src = (unsigned long long)(const void*)(sboxes + r);
        asm volatile("global_load_async_to_lds_b128 %0, %1, off"
                     :: "v"(dst), "v"(src) : "memory");
    }
    if (tid < NW) keepw[tid] = 0xFFFFFFFFu;
    asm volatile("s_wait_asynccnt 0" ::: "memory");
    __syncthreads();

    for (int w = 0; w < NW; ++w) {
        // ---- (a) diagonal: wave 0, serial greedy over the 32-box window ----
        if (wave == 0) {
            const int i = (w << 5) + lane;
            const bool valid = i < M_PROP;
            bool alive = valid && ((keepw[w] >> lane) & 1u);
            float4 b = sbox[valid ? i : 0];
            float area = (b.z - b.x) * (b.w - b.y);
            unsigned aliveMask = (unsigned)__ballot(alive);
            for (int t = 0; t < 32; ++t) {
                if (!((aliveMask >> t) & 1u)) continue;       // uniform in wave
                float lx1 = __shfl(b.x, t), ly1 = __shfl(b.y, t);
                float lx2 = __shfl(b.z, t), ly2 = __shfl(b.w, t);
                float la  = __shfl(area, t);
                if (alive && lane > t) {
                    if (iou_f(lx1, ly1, lx2, ly2, la, b) > NMS_THR) alive = false;
                }
                aliveMask = (unsigned)__ballot(alive);
            }
            if (alive) {                                       // compact leaders
                int pos = __popc(aliveMask & ((1u << lane) - 1u));
                lbox[pos] = b;
                larea[pos] = area;
            }
            if (lane == 0) {
                keepw[w] = aliveMask;
                misc[0] = (unsigned)__popc(aliveMask);
            }
        }
        __syncthreads();

        // ---- (b) off-diagonal: each wave owns one later keep word ----------
        const int L = (int)misc[0];
        if (L > 0) {
            for (int word = w + 1 + wave; word < NW; word += 32) {
                const unsigned old = keepw[word];              // LDS broadcast
                const int j = (word << 5) + lane;
                bool sup = false;
                if (j < M_PROP && ((old >> lane) & 1u)) {
                    const float4 bj = sbox[j];
                    const float bjx1 = bj.x, bjy1 = bj.y, bjx2 = bj.z, bjy2 = bj.w;
                    const float areaj = (bjx2 - bjx1) * (bjy2 - bjy1);
                    for (int l = 0; l < L; ++l) {
                        const float4 bl = lbox[l];             // LDS broadcast
                        float iw = fminf(bjx2, bl.z) - fmaxf(bjx1, bl.x);
                        float ih = fminf(bjy2, bl.w) - fmaxf(bjy1, bl.y);
                        iw = fmaxf(iw, 0.0f); ih = fmaxf(ih, 0.0f);
                        float inter = iw * ih;
                        float iou = inter / (areaj + larea[l] - inter + 1e-8f);
                        sup = sup || (iou > NMS_THR);
                    }
                }
                unsigned supm = (unsigned)__ballot(sup);
                if (lane == 0) keepw[word] = old & ~supm;      // sole writer
            }
        }
        __syncthreads();
    }

    // ---- Stable partition ranks: kept (in order) first, suppressed after ---
    if (tid < NW) {
        unsigned m = keepw[tid];
        if (tid == NW - 1) { m &= TAILMASK; keepw[tid] = m; }
        wsum[tid] = __popc(m);
    }
    __syncthreads();
    if (tid == 0) {
        unsigned s = 0;
        for (int v = 0; v < NW; ++v) { unsigned c = wsum[v]; wsum[v] = s; s += c; }
        wsum[NW] = s;
    }
    __syncthreads();
    const unsigned K = wsum[NW];

    if (tid < NW) {
        unsigned m = keepw[tid];
        unsigned kbase = wsum[tid];
        unsigned sbase = K + ((unsigned)tid * 32u - kbase);
        unsigned kc = 0, sc = 0;
        for (int b = 0; b < 32; ++b) {
            int i = tid * 32 + b;
            if (i >= M_PROP) break;
            bool kept = (m >> b) & 1u;
            unsigned rank = kept ? (kbase + kc) : (sbase + sc);
            if (kept) ++kc; else ++sc;
            if (rank < TOPK_OUT) {
                unsigned oi = order[i];
                const float* p = prop + (size_t)oi * 6;
                float* o = out + (size_t)rank * 5;
                o[0] = p[0]; o[1] = p[1]; o[2] = p[2]; o[3] = p[3]; o[4] = p[4];
            }
        }
    }
}

extern "C" void kernel_launch(void* const* d_in, const int* in_sizes, int n_in,
                              void* d_out, int out_size, void* d_ws, size_t ws_size,
                              hipStream_t stream) {
    (void)in_sizes; (void)n_in; (void)out_size; (void)ws_size;
    const float* prop = (const float*)d_in[0];
    float* out = (float*)d_out;
    float4*   sboxes = (float4*)d_ws;
    unsigned* order  = (unsigned*)((char*)d_ws + WS_ORDER_OFF);

    rpn_sort_kernel<<<1, NTHREADS, NSORT * sizeof(unsigned long long), stream>>>(
        prop, sboxes, order);

    rpn_nms_kernel<<<1, NTHREADS, L_TOTAL, stream>>>(prop, sboxes, order, out);
}